// BaseLauxModel_69741678952701
// MI455X (gfx1250) — compile-verified
//
#include <hip/hip_runtime.h>
#include <hip/hip_bf16.h>

#define S 8192
#define E 64
#define C 256

typedef __attribute__((ext_vector_type(4))) float f4;
typedef __attribute__((ext_vector_type(2))) float v2f;
typedef __attribute__((ext_vector_type(8))) float v8f;

// ---------------------------------------------------------------------------
// Kernel 1: zero-fill the 512MB dense output with NT b128 stores.
// Also zeroes the 128-float reduction workspace and the single tail element.
// ---------------------------------------------------------------------------
__global__ void zero_fill_kernel(float* __restrict__ out, float* __restrict__ red) {
  const long long n4 = ((long long)S * E * C + 1) / 4;  // 33554432 float4 -> covers [0, 134217728)
  long long gid    = (long long)blockIdx.x * blockDim.x + threadIdx.x;
  long long stride = (long long)gridDim.x * blockDim.x;
  if (gid < 2 * E) red[gid] = 0.0f;
  if (gid == 2 * E) out[(long long)S * E * C] = 0.0f;  // last element (index 134217728)
  f4* out4 = (f4*)out;
  f4 z = {0.f, 0.f, 0.f, 0.f};
  for (long long i = gid; i < n4; i += stride)
    __builtin_nontemporal_store(z, out4 + i);
}

// ---------------------------------------------------------------------------
// Kernel 2: one thread per token. Softmax over 64 experts, top-2 gate weights,
// one-hot -> index via exact f32 iota dot products, scatter 2 nonzeros.
// Also stores the full softmax row to workspace for the WMMA reduction.
// ---------------------------------------------------------------------------
__global__ void gating_kernel(const float* __restrict__ logits,
                              const float* __restrict__ mask1,
                              const float* __restrict__ mask2,
                              const float* __restrict__ loc1,
                              const float* __restrict__ loc2,
                              float* __restrict__ out,
                              float* __restrict__ gates_ws) {
  int s = blockIdx.x * blockDim.x + threadIdx.x;
  if (s >= S) return;

  const f4* lg = (const f4*)(logits + (long long)s * E);
  float l[E];
  float mx = -3.402823466e38f;
#pragma unroll
  for (int i = 0; i < E / 4; ++i) {
    f4 v = lg[i];
    l[4 * i + 0] = v.x; l[4 * i + 1] = v.y; l[4 * i + 2] = v.z; l[4 * i + 3] = v.w;
    mx = fmaxf(mx, fmaxf(fmaxf(v.x, v.y), fmaxf(v.z, v.w)));
  }
  float sum = 0.f;
#pragma unroll
  for (int e = 0; e < E; ++e) { l[e] = __expf(l[e] - mx); sum += l[e]; }
  float inv = 1.f / sum;

  const f4* m1p = (const f4*)(mask1 + (long long)s * E);
  const f4* m2p = (const f4*)(mask2 + (long long)s * E);
  f4* gout = (f4*)(gates_ws + (long long)s * E);
  float d1 = 0.f, d2 = 0.f, e1f = 0.f, e2f = 0.f;
#pragma unroll
  for (int i = 0; i < E / 4; ++i) {
    f4 a = m1p[i], b = m2p[i];
    f4 g;
    g.x = l[4 * i + 0] * inv; g.y = l[4 * i + 1] * inv;
    g.z = l[4 * i + 2] * inv; g.w = l[4 * i + 3] * inv;
    gout[i] = g;
    d1 += g.x * a.x + g.y * a.y + g.z * a.z + g.w * a.w;
    d2 += g.x * b.x + g.y * b.y + g.z * b.z + g.w * b.w;
    e1f += a.x * (float)(4 * i) + a.y * (float)(4 * i + 1) + a.z * (float)(4 * i + 2) + a.w * (float)(4 * i + 3);
    e2f += b.x * (float)(4 * i) + b.y * (float)(4 * i + 1) + b.z * (float)(4 * i + 2) + b.w * (float)(4 * i + 3);
  }

  const f4* L1 = (const f4*)(loc1 + (long long)s * C);
  const f4* L2 = (const f4*)(loc2 + (long long)s * C);
  float c1f = 0.f, c2f = 0.f, l1v = 0.f, l2v = 0.f;
#pragma unroll 8
  for (int i = 0; i < C / 4; ++i) {
    f4 u = L1[i], w = L2[i];
    c1f += u.x * (float)(4 * i) + u.y * (float)(4 * i + 1) + u.z * (float)(4 * i + 2) + u.w * (float)(4 * i + 3);
    c2f += w.x * (float)(4 * i) + w.y * (float)(4 * i + 1) + w.z * (float)(4 * i + 2) + w.w * (float)(4 * i + 3);
    l1v += u.x + u.y + u.z + u.w;   // == 1.0 (one-hot), kept for exact reference semantics
    l2v += w.x + w.y + w.z + w.w;
  }

  int e1 = (int)(e1f + 0.5f), e2 = (int)(e2f + 0.5f);
  int c1 = (int)(c1f + 0.5f), c2 = (int)(c2f + 0.5f);
  float denom = fmaxf(d1 + d2, 1.1920929e-07f);  // FLT_EPSILON, as in reference
  float g1 = (d1 / denom) * l1v;
  float g2 = (d2 / denom) * l2v;
  out[1 + ((long long)s * E + e1) * C + c1] = g1;
  out[1 + ((long long)s * E + e2) * C + c2] = g2;
}

// ---------------------------------------------------------------------------
// Kernel 3: column sums over tokens via V_WMMA_F32_16X16X4_F32.
//   me_sum[e] = sum_s gates[s,e],  ce_sum[e] = sum_s mask1[s,e]
// A = ones(16x4) so C[m,n] = sum_k B[k,n]; experts ride the N (lane) axis,
// tokens ride K. Result read from C row M=0 (VGPR 0, lanes 0..15).
// 32 waves x 256 tokens each; partials merged with global f32 atomics.
// ---------------------------------------------------------------------------
__global__ void moe_reduce_wmma(const float* __restrict__ gates,
                                const float* __restrict__ mask1,
                                float* __restrict__ red) {
  int wave = (blockIdx.x * blockDim.x + threadIdx.x) >> 5;
  int lane = threadIdx.x & 31;
  int half = lane >> 4;
  int col  = lane & 15;

  v2f ones; ones.x = 1.f; ones.y = 1.f;
  v8f accG[4] = {};
  v8f accM[4] = {};

  const int tokensPerWave = S / 32;   // 256
  int t0 = wave * tokensPerWave;
  for (int t = t0; t < t0 + tokensPerWave; t += 4) {
    int r0 = t + 2 * half;            // lanes 0-15: tokens t,t+1 ; lanes 16-31: t+2,t+3
    const float* g0 = gates + (long long)r0 * E;
    const float* g1 = g0 + E;
    const float* m0 = mask1 + (long long)r0 * E;
    const float* m1 = m0 + E;
#pragma unroll
    for (int eb = 0; eb < 4; ++eb) {
      v2f bG; bG.x = g0[eb * 16 + col]; bG.y = g1[eb * 16 + col];
      v2f bM; bM.x = m0[eb * 16 + col]; bM.y = m1[eb * 16 + col];
      accG[eb] = __builtin_amdgcn_wmma_f32_16x16x4_f32(false, ones, false, bG,
                                                       (short)0, accG[eb], false, false);
      accM[eb] = __builtin_amdgcn_wmma_f32_16x16x4_f32(false, ones, false, bM,
                                                       (short)0, accM[eb], false, false);
    }
  }

  if (lane < 16) {
#pragma unroll
    for (int eb = 0; eb < 4; ++eb) {
      atomicAdd(&red[eb * 16 + col],     accG[eb][0]);  // me_sum partial
      atomicAdd(&red[E + eb * 16 + col], accM[eb][0]);  // ce_sum partial
    }
  }
}

// ---------------------------------------------------------------------------
// Kernel 4: l_aux = (E / S^2) * sum_e me_sum[e] * ce_sum[e]  -> out[0]
// ---------------------------------------------------------------------------
__global__ void laux_kernel(const float* __restrict__ red, float* __restrict__ out) {
  int lane = threadIdx.x & 31;
  float v = red[lane] * red[E + lane] + red[lane + 32] * red[E + lane + 32];
#pragma unroll
  for (int off = 16; off > 0; off >>= 1)
    v += __shfl_xor(v, off, 32);
  if (lane == 0)
    out[0] = v * ((float)E / ((float)S * (float)S));
}

extern "C" void kernel_launch(void* const* d_in, const int* in_sizes, int n_in,
                              void* d_out, int out_size, void* d_ws, size_t ws_size,
                              hipStream_t stream) {
  const float* logits = (const float*)d_in[0];
  const float* mask1  = (const float*)d_in[1];
  const float* mask2  = (const float*)d_in[2];
  const float* loc1   = (const float*)d_in[3];
  const float* loc2   = (const float*)d_in[4];
  float* out = (float*)d_out;

  float* gates_ws = (float*)d_ws;                 // [S*E] floats (2 MB)
  float* red      = gates_ws + (size_t)S * E;     // [2*E] floats (me_sum | ce_sum)

  // 1) 512MB NT zero-fill (store-bandwidth bound: ~22us floor at 23.3 TB/s)
  zero_fill_kernel<<<8192, 256, 0, stream>>>(out, red);
  // 2) per-token gating + 2-element scatter (runs after fill; same stream)
  gating_kernel<<<S / 256, 256, 0, stream>>>(logits, mask1, mask2, loc1, loc2, out, gates_ws);
  // 3) WMMA f32 column-sum reduction over tokens
  moe_reduce_wmma<<<4, 256, 0, stream>>>(gates_ws, mask1, red);
  // 4) scalar l_aux
  laux_kernel<<<1, 32, 0, stream>>>(red, out);
}